// NCECriterion_23046794510890
// MI455X (gfx1250) — compile-verified
//
#include <hip/hip_runtime.h>
#include <hip/hip_bf16.h>

typedef __attribute__((ext_vector_type(2))) float v2f;
typedef __attribute__((ext_vector_type(8))) float v8f;

#define NCE_K   256
#define NCE_T   0.07f
#define NCE_EPS 1e-7f
#define D_DIM   128
#define B_N     4096
#define MEM_N   1000000

// One block (8 wave32) per example b. Each wave computes logits for 32 k's
// (two 16-row WMMA groups) using V_WMMA_F32_16X16X4_F32 with x[b] replicated
// across all 16 B-matrix columns (GEMV-as-WMMA; memory-bound anyway).
__global__ __launch_bounds__(256) void nce_main_kernel(
    const float* __restrict__ x,
    const int*   __restrict__ targets,
    const float* __restrict__ memory,
    const int*   __restrict__ indices,
    float*       __restrict__ partial)
{
    __shared__ float lds_x[D_DIM];
    __shared__ float lds_logit[NCE_K];
    __shared__ float lds_red[NCE_K];

    const int b    = blockIdx.x;
    const int t    = threadIdx.x;
    const int lane = t & 31;
    const int wave = t >> 5;

    // Stage x[b, :] into LDS (128 floats).
    if (t < D_DIM) lds_x[t] = x[b * D_DIM + t];
    __syncthreads();

    // A-matrix (16x4 f32) layout: lane l -> row M = l%16;
    // low lanes hold K-slice {0,1}, high lanes {2,3}.
    // B-matrix (4x16 f32) layout mirrors this in K, column = l%16.
    const int mrow  = lane & 15;
    const int half  = lane >> 4;   // 0 = lanes 0-15, 1 = lanes 16-31
    const int dbase = half * 2;    // per-lane offset within each 4-wide D slice

    for (int g = 0; g < 2; ++g) {
        const int kbase = wave * 32 + g * 16;
        const int kk    = kbase + mrow;
        // nce_average: column 0 replaced by the positive (target) index
        const int row_idx = (kk == 0) ? targets[b] : indices[b * NCE_K + kk];
        const float* wrow = memory + (long long)row_idx * D_DIM;

        v8f c = {};
#pragma unroll
        for (int d0 = 0; d0 < D_DIM; d0 += 4) {
            v2f a, bb;
            a.x  = wrow[d0 + dbase];
            a.y  = wrow[d0 + dbase + 1];
            bb.x = lds_x[d0 + dbase];
            bb.y = lds_x[d0 + dbase + 1];
            // 8 args: (neg_a, A, neg_b, B, c_mod, C, reuse_a, reuse_b)
            c = __builtin_amdgcn_wmma_f32_16x16x4_f32(
                    false, a, false, bb, (short)0, c, false, false);
        }

        // Column N==0 lanes (lane 0 and lane 16) hold the 16 valid dot
        // products: lane 0 -> rows 0..7 in c[0..7], lane 16 -> rows 8..15.
        if (mrow == 0) {
#pragma unroll
            for (int v = 0; v < 8; ++v)
                lds_logit[kbase + half * 8 + v] = c[v];
        }
    }
    __syncthreads();

    // sims = exp(logit / T); normalize over K; scale by K/N (matches reference,
    // including no max-subtraction).
    const float invT = 1.0f / NCE_T;
    const float e = expf(lds_logit[t] * invT);
    lds_red[t] = e;
    __syncthreads();
#pragma unroll
    for (int s = 128; s > 0; s >>= 1) {
        if (t < s) lds_red[t] += lds_red[t + s];
        __syncthreads();
    }
    const float S = lds_red[0];

    const float scale = (float)NCE_K / (float)MEM_N;
    const float noise = (float)(NCE_K - 1) / (float)MEM_N;
    const float sim   = e / S * scale;

    float term;
    if (t == 0) term = logf(sim   / (sim + noise + NCE_EPS));  // log lnPmt
    else        term = logf(noise / (sim + noise + NCE_EPS));  // log lnPon
    __syncthreads();               // all reads of lds_logit are done above
    lds_logit[t] = term;
    __syncthreads();
#pragma unroll
    for (int s = 128; s > 0; s >>= 1) {
        if (t < s) lds_logit[t] += lds_logit[t + s];
        __syncthreads();
    }
    if (t == 0) partial[b] = lds_logit[0];
}

__global__ __launch_bounds__(256) void nce_finalize_kernel(
    const float* __restrict__ partial, float* __restrict__ out)
{
    __shared__ float red[256];
    const int t = threadIdx.x;
    float s = 0.0f;
    for (int i = t; i < B_N; i += 256) s += partial[i];
    red[t] = s;
    __syncthreads();
#pragma unroll
    for (int w = 128; w > 0; w >>= 1) {
        if (t < w) red[t] += red[t + w];
        __syncthreads();
    }
    if (t == 0) out[0] = -red[0] / (float)B_N;
}

extern "C" void kernel_launch(void* const* d_in, const int* in_sizes, int n_in,
                              void* d_out, int out_size, void* d_ws, size_t ws_size,
                              hipStream_t stream) {
    const float* x       = (const float*)d_in[0];  // [B, D]
    const int*   targets = (const int*)  d_in[1];  // [B]
    const float* memory  = (const float*)d_in[2];  // [N, D]
    const int*   indices = (const int*)  d_in[3];  // [B, K]
    float* partial = (float*)d_ws;                 // [B] scratch
    float* out     = (float*)d_out;                // scalar

    nce_main_kernel<<<B_N, 256, 0, stream>>>(x, targets, memory, indices, partial);
    nce_finalize_kernel<<<1, 256, 0, stream>>>(partial, out);
}